// multiply_57200374448374
// MI455X (gfx1250) — compile-verified
//
#include <hip/hip_runtime.h>

typedef __attribute__((ext_vector_type(2))) float v2f;
typedef __attribute__((ext_vector_type(8))) float v8f;

// Problem constants: B=1024, ele = 32*16 = 512, npairs = 512*513/2 = 131328.
// Pair space tiled as 32x32 grid of 16x16 tiles; 496 strictly-upper tiles +
// 32 diagonal tiles. Packed triu index: idx(i,j) = i*(1025-i)/2 + (j-i),
// with row-to-row recurrence  idx(i+1,j) - idx(i,j) = 511 - i.
#define ELE      512
#define NPAIRS   131328

__global__ __launch_bounds__(256) void triu_outer_wmma(const float* __restrict__ X,
                                                       float* __restrict__ out) {
    __shared__ float xs[ELE];

    const int b   = blockIdx.x;
    const int tid = threadIdx.x;

    // Stage this batch row (2 KB) into LDS, coalesced.
    const float* __restrict__ xrow = X + (size_t)b * ELE;
    xs[tid]       = xrow[tid];
    xs[tid + 256] = xrow[tid + 256];
    __syncthreads();

    const int  wave    = tid >> 5;        // 8 waves per block (wave32)
    const int  lane    = tid & 31;
    const int  lane15  = lane & 15;
    const int  half    = lane >> 4;       // 0: rows i0+r, 1: rows i0+8+r
    const bool lowHalf = (half == 0);
    float* __restrict__ outB = out + (size_t)b * (size_t)NPAIRS;

    // ---- Off-diagonal tiles (J > I): unconditional, fully coalesced stores.
    for (int I = 0; I < 31; ++I) {
        // A fragment for this tile-row, loaded once: A[m,0] = x[I*16+m].
        // Load unconditionally (address valid for all lanes), mask with a
        // plain select so the compiler emits v_cndmask, not a predicated load.
        const float av = xs[I * 16 + lane15];
        v2f a;
        a.x = lowHalf ? av : 0.0f;
        a.y = 0.0f;

        const int      i0   = I * 16 + half * 8;                 // first row this lane stores
        const unsigned off0 = (unsigned)(i0 * (1025 - i0)) >> 1; // triu offset(i0)

        for (int J = I + 1 + wave; J < 32; J += 8) {             // wave-uniform trip count
            const float bv = xs[J * 16 + lane15];                // unconditional ds_load
            v2f bm;
            bm.x = lowHalf ? bv : 0.0f;                          // v_cndmask
            bm.y = 0.0f;

            v8f c = {};
            c = __builtin_amdgcn_wmma_f32_16x16x4_f32(
                    false, a, false, bm, (short)0, c, false, false);

            // C layout: VGPR r -> row i0 + r, col j = J*16 + (lane&15).
            const int j    = J * 16 + lane15;
            unsigned  idx  = off0 + (unsigned)(j - i0);
            unsigned  step = 511u - (unsigned)i0;
#pragma unroll
            for (int r = 0; r < 8; ++r) {
                outB[idx] = c[r];      // j > i always here: no predicate
                idx += step;
                --step;
            }
        }
    }

    // ---- Diagonal tiles (J == I): A == B, predicated stores (j >= i).
    for (int I = wave; I < 32; I += 8) {
        const float av = xs[I * 16 + lane15];
        v2f a;
        a.x = lowHalf ? av : 0.0f;
        a.y = 0.0f;

        v8f c = {};
        c = __builtin_amdgcn_wmma_f32_16x16x4_f32(
                false, a, false, a, (short)0, c, false, false);

        const int      i0   = I * 16 + half * 8;
        const unsigned off0 = (unsigned)(i0 * (1025 - i0)) >> 1;
        const int      d    = lane15 - half * 8;  // j - i0 (may be negative)
        unsigned       idx  = off0 + (unsigned)d; // only consumed when d >= r
        unsigned       step = 511u - (unsigned)i0;
#pragma unroll
        for (int r = 0; r < 8; ++r) {
            if (d >= r) outB[idx] = c[r];
            idx += step;
            --step;
        }
    }
}

extern "C" void kernel_launch(void* const* d_in, const int* in_sizes, int n_in,
                              void* d_out, int out_size, void* d_ws, size_t ws_size,
                              hipStream_t stream) {
    const float* X = (const float*)d_in[0];   // [1024, 32, 16] fp32
    float* out     = (float*)d_out;           // [1024, 131328, 1] fp32
    (void)in_sizes; (void)n_in; (void)out_size; (void)d_ws; (void)ws_size;

    triu_outer_wmma<<<1024, 256, 0, stream>>>(X, out);
}